// SelfAttention_3375844294964
// MI455X (gfx1250) — compile-verified
//
#include <hip/hip_runtime.h>
#include <math.h>

// ---------------- CDNA5 (gfx1250) wave32 WMMA f32 16x16x4 ----------------
typedef __attribute__((ext_vector_type(2))) float v2f;
typedef __attribute__((ext_vector_type(8))) float v8f;

#define WMMA_F32X4(a, b, c) \
  __builtin_amdgcn_wmma_f32_16x16x4_f32(false, (a), false, (b), (short)0, (c), false, false)

// Problem constants (match reference)
constexpr int BATCH = 32;
constexpr int DIM   = 4096;
constexpr int NH    = 32;
constexpr int NKV   = 8;
constexpr int NREP  = 4;
constexpr int HD    = 128;
constexpr int TMAX  = 2048;
constexpr int START = 2047;   // start_pos in the reference
constexpr int KVROW = NKV * HD;  // 1024 floats per cache time-step

// -------------------------------------------------------------------------
// GEMM: Y[M x N] = X[M x K] * W[N x K]^T    (M = 32, grid.y = 2 tiles of 16)
// One wave per 16x16 output tile; K-loop in steps of 8 with 2 accumulators.
// Weights are single-use -> non-temporal loads (don't pollute L2 vs KV cache).
// -------------------------------------------------------------------------
__global__ __launch_bounds__(256)
void gemm_xwT_wmma(const float* __restrict__ X, const float* __restrict__ W,
                   float* __restrict__ Y, int N, int K) {
  const int lane = threadIdx.x & 31;
  const int wave = threadIdx.x >> 5;
  const int n0   = (blockIdx.x * 8 + wave) << 4;
  const int m0   = blockIdx.y << 4;
  const int row  = lane & 15;                // A-row / B-col within tile
  const int koff = (lane < 16) ? 0 : 2;      // K sub-offset per half-wave

  const float* xr = X + (size_t)(m0 + row) * K + koff;
  const float* wr = W + (size_t)(n0 + row) * K + koff;

  v8f c0 = {0,0,0,0,0,0,0,0};
  v8f c1 = {0,0,0,0,0,0,0,0};

#pragma unroll 4
  for (int k = 0; k < K; k += 8) {
    v2f a0 = *(const v2f*)(xr + k);
    v2f a1 = *(const v2f*)(xr + k + 4);
    v2f b0 = __builtin_nontemporal_load((const v2f*)(wr + k));
    v2f b1 = __builtin_nontemporal_load((const v2f*)(wr + k + 4));
    c0 = WMMA_F32X4(a0, b0, c0);
    c1 = WMMA_F32X4(a1, b1, c1);
  }

  v8f cs = c0 + c1;
  const int mb = (lane < 16) ? 0 : 8;        // C layout: row = v + mb
#pragma unroll
  for (int v = 0; v < 8; ++v)
    Y[(size_t)(m0 + mb + v) * N + n0 + row] = cs[v];
}

// -------------------------------------------------------------------------
// RoPE on xq (32 heads) and xk (8 heads), single position, in workspace.
// One thread per (b, head, pair). 32*40*64 = 81920 threads exactly.
// -------------------------------------------------------------------------
__global__ __launch_bounds__(256)
void rope_kernel(float* __restrict__ xq, float* __restrict__ xk,
                 const float* __restrict__ cosv, const float* __restrict__ sinv) {
  const int idx  = blockIdx.x * 256 + threadIdx.x;
  const int pair = idx & 63;
  const int head = (idx >> 6) % (NH + NKV);
  const int b    = idx / (64 * (NH + NKV));
  const float c  = cosv[pair];
  const float sn = sinv[pair];
  float* p = (head < NH)
      ? (xq + ((size_t)b * NH + head) * HD + pair * 2)
      : (xk + ((size_t)b * NKV + (head - NH)) * HD + pair * 2);
  const float x0 = p[0], x1 = p[1];
  p[0] = x0 * c - x1 * sn;
  p[1] = x0 * sn + x1 * c;
}

// -------------------------------------------------------------------------
// Decode attention for one (batch, kv-head) per workgroup (grid 8 x 32).
// 8 waves: each wave scores 256 keys via WMMA (A = 4 q-reps padded to 16
// rows), scores -> LDS, block-wide softmax (max & sum trees), then each wave
// owns a 16-wide hd tile and accumulates P*V over all 2048 keys via WMMA.
// Key/value row t==2047 is read from workspace (inputs are never mutated).
// KV cache is single-use -> non-temporal loads.
// -------------------------------------------------------------------------
__global__ __launch_bounds__(256)
void attn_decode_wmma(const float* __restrict__ xq, const float* __restrict__ xk,
                      const float* __restrict__ xv, const float* __restrict__ ck,
                      const float* __restrict__ cv, float* __restrict__ out) {
  const int kv   = blockIdx.x;          // 0..7
  const int b    = blockIdx.y;          // 0..31
  const int tid  = threadIdx.x;
  const int lane = tid & 31;
  const int wave = tid >> 5;
  const int col  = lane & 15;
  const int koff = (lane < 16) ? 0 : 2;

  __shared__ float s[NREP][TMAX];       // 32 KB score buffer
  __shared__ float red[4 * 256];        // reduction scratch

  const float scale = 0.08838834764831845f;  // 1/sqrt(128)
  const float* qb    = xq + ((size_t)b * NH + kv * NREP) * HD;
  const float* kbase = ck + (size_t)b * TMAX * KVROW + kv * HD;
  const float* knew  = xk + ((size_t)b * NKV + kv) * HD;

  // ---- Phase 1: scores for this wave's 256 keys -------------------------
  const int tbase = wave * 256;
  for (int tt = 0; tt < 256; tt += 16) {
    const int tg = tbase + tt + col;    // this lane's key row for B
    const float* kp = (tg == START) ? knew : (kbase + (size_t)tg * KVROW);
    v8f c0 = {0,0,0,0,0,0,0,0};
    v8f c1 = {0,0,0,0,0,0,0,0};
#pragma unroll 4
    for (int h = 0; h < HD; h += 8) {
      v2f a0 = {0, 0}, a1 = {0, 0};
      if (col < NREP) {                 // rows 4..15 of A are zero padding
        a0 = *(const v2f*)(qb + col * HD + h + koff);
        a1 = *(const v2f*)(qb + col * HD + h + 4 + koff);
      }
      v2f b0 = __builtin_nontemporal_load((const v2f*)(kp + h + koff));
      v2f b1 = __builtin_nontemporal_load((const v2f*)(kp + h + 4 + koff));
      c0 = WMMA_F32X4(a0, b0, c0);
      c1 = WMMA_F32X4(a1, b1, c1);
    }
    v8f cs = c0 + c1;
    if (lane < 16) {                    // C rows 0..3 = the 4 valid reps
#pragma unroll
      for (int r = 0; r < NREP; ++r)
        s[r][tbase + tt + lane] = cs[r] * scale;
    }
  }
  __syncthreads();

  // ---- Phase 2: block softmax (max, then exp+sum) -----------------------
  float pm[NREP] = {-1e30f, -1e30f, -1e30f, -1e30f};
  for (int t = tid; t < TMAX; t += 256)
#pragma unroll
    for (int r = 0; r < NREP; ++r) pm[r] = fmaxf(pm[r], s[r][t]);
#pragma unroll
  for (int r = 0; r < NREP; ++r) red[r * 256 + tid] = pm[r];
  __syncthreads();
  for (int st = 128; st > 0; st >>= 1) {
    if (tid < st)
#pragma unroll
      for (int r = 0; r < NREP; ++r)
        red[r * 256 + tid] = fmaxf(red[r * 256 + tid], red[r * 256 + tid + st]);
    __syncthreads();
  }
  float gmax[NREP];
#pragma unroll
  for (int r = 0; r < NREP; ++r) gmax[r] = red[r * 256];
  __syncthreads();

  float ps[NREP] = {0.f, 0.f, 0.f, 0.f};
  for (int t = tid; t < TMAX; t += 256)
#pragma unroll
    for (int r = 0; r < NREP; ++r) {
      const float e = __expf(s[r][t] - gmax[r]);
      s[r][t] = e;
      ps[r] += e;
    }
#pragma unroll
  for (int r = 0; r < NREP; ++r) red[r * 256 + tid] = ps[r];
  __syncthreads();
  for (int st = 128; st > 0; st >>= 1) {
    if (tid < st)
#pragma unroll
      for (int r = 0; r < NREP; ++r)
        red[r * 256 + tid] += red[r * 256 + tid + st];
    __syncthreads();
  }
  float gsum[NREP];
#pragma unroll
  for (int r = 0; r < NREP; ++r) gsum[r] = red[r * 256];

  // ---- Phase 3: P*V, each wave owns a 16-wide hd tile -------------------
  const int hd0 = wave * 16;
  const float* vcol = cv + (size_t)b * TMAX * KVROW + kv * HD + hd0 + col;
  const float* vnew = xv + ((size_t)b * NKV + kv) * HD + hd0 + col;

  v8f o0 = {0,0,0,0,0,0,0,0};
  v8f o1 = {0,0,0,0,0,0,0,0};
#pragma unroll 2
  for (int t0 = 0; t0 < TMAX; t0 += 8) {
    const int ta = t0 + koff;
    v2f a0 = {0, 0}, a1 = {0, 0};
    if (col < NREP) {                   // P rows 4..15 are zero padding
      a0 = *(const v2f*)(&s[col][ta]);
      a1 = *(const v2f*)(&s[col][ta + 4]);
    }
    v2f b0, b1;
    b0.x = __builtin_nontemporal_load(vcol + (size_t)ta * KVROW);
    b0.y = (ta + 1 == START) ? *vnew
           : __builtin_nontemporal_load(vcol + (size_t)(ta + 1) * KVROW);
    b1.x = (ta + 4 == START) ? *vnew
           : __builtin_nontemporal_load(vcol + (size_t)(ta + 4) * KVROW);
    b1.y = (ta + 5 == START) ? *vnew
           : __builtin_nontemporal_load(vcol + (size_t)(ta + 5) * KVROW);
    o0 = WMMA_F32X4(a0, b0, o0);
    o1 = WMMA_F32X4(a1, b1, o1);
  }

  v8f os = o0 + o1;
  if (lane < 16) {                      // C rows 0..3 = the 4 reps
#pragma unroll
    for (int r = 0; r < NREP; ++r)
      out[((size_t)b * NH + kv * NREP + r) * HD + hd0 + col] = os[r] / gsum[r];
  }
}

// -------------------------------------------------------------------------
extern "C" void kernel_launch(void* const* d_in, const int* in_sizes, int n_in,
                              void* d_out, int out_size, void* d_ws, size_t ws_size,
                              hipStream_t stream) {
  const float* x  = (const float*)d_in[0];
  const float* fc = (const float*)d_in[1];
  const float* fs = (const float*)d_in[2];
  const float* wq = (const float*)d_in[3];
  const float* wk = (const float*)d_in[4];
  const float* wv = (const float*)d_in[5];
  const float* wv_ = wv;
  const float* wo = (const float*)d_in[6];
  const float* ck = (const float*)d_in[7];
  const float* cv = (const float*)d_in[8];
  (void)in_sizes; (void)n_in; (void)out_size; (void)ws_size;

  float* ws = (float*)d_ws;
  float* xq = ws;                        // 32*4096
  float* xk = xq + BATCH * DIM;          // 32*1024
  float* xv = xk + BATCH * NKV * HD;     // 32*1024
  float* ao = xv + BATCH * NKV * HD;     // 32*4096

  // QKV projections (x @ w.T), M=32 -> grid.y = 2 tiles of 16 rows
  gemm_xwT_wmma<<<dim3(DIM / 128, 2), 256, 0, stream>>>(x, wq, xq, DIM, DIM);
  gemm_xwT_wmma<<<dim3((NKV * HD) / 128, 2), 256, 0, stream>>>(x, wk, xk, NKV * HD, DIM);
  gemm_xwT_wmma<<<dim3((NKV * HD) / 128, 2), 256, 0, stream>>>(x, wv_, xv, NKV * HD, DIM);

  // RoPE on q and k (in workspace; inputs untouched)
  rope_kernel<<<(BATCH * (NH + NKV) * 64) / 256, 256, 0, stream>>>(xq, xk, fc, fs);

  // GQA decode attention: one WG per (kv-head, batch)
  attn_decode_wmma<<<dim3(NKV, BATCH), 256, 0, stream>>>(xq, xk, xv, ck, cv, ao);

  // Output projection -> d_out
  gemm_xwT_wmma<<<dim3(DIM / 128, 2), 256, 0, stream>>>(ao, wo, (float*)d_out, DIM, DIM);
}